// multiplication_20864951124494
// MI455X (gfx1250) — compile-verified
//
#include <hip/hip_runtime.h>
#include <stdint.h>

typedef float v2f __attribute__((ext_vector_type(2)));
typedef float v8f __attribute__((ext_vector_type(8)));

#define BATCH 4
#define HW    4096
#define DIM   64
#define CH    256
#define WAVES 4
#define ROWS_PER_WG (16 * WAVES)
#define NTILES (HW / 16)

// One 16-byte async global->LDS transfer per lane (CDNA5 GLOBAL_LOAD_ASYNC_TO_LDS_B128,
// tracked by ASYNCcnt; LDS byte address = low 32 bits of the flat shared-pointer).
__device__ __forceinline__ void async_copy16(uint32_t lds_byte, const void* gsrc) {
    unsigned long long ga = (unsigned long long)(uintptr_t)gsrc;
    asm volatile("global_load_async_to_lds_b128 %0, %1, off"
                 :: "v"(lds_byte), "v"(ga)
                 : "memory");
}

// Flash-attention pass: each wave owns 16 output rows (i), loops over all 4096
// j in 16-row tiles; tiles are double-buffered in LDS via async-to-LDS DMA and
// shared by the 4 waves of the workgroup.
__global__ __launch_bounds__(32 * WAVES) void attn_flash_kernel(
    const float* __restrict__ xq,   // x_query (B,HW,D)
    const float* __restrict__ pg,   // phi_g   (B,HW,D)
    const float* __restrict__ xv,   // x_value (B,HW,C)
    const float* __restrict__ xres, // x       (B,HW,C)
    const float* __restrict__ kp,   // scalar k
    float* __restrict__ out)        // (B,HW,2C)
{
    __shared__ __align__(16) float qs[2][16 * DIM];  // 2 x 4 KB  q tiles
    __shared__ __align__(16) float vs[2][16 * CH];   // 2 x 16 KB V tiles

    const int b    = blockIdx.y;
    const int tid  = threadIdx.x;
    const int lane = tid & 31;
    const int wave = tid >> 5;
    const int lp   = lane & 15;   // low lane index: i (cols of T) / c
    const int half = lane >> 4;   // half-wave selector
    const int i0   = blockIdx.x * ROWS_PER_WG + wave * 16;

    const float kk = kp[0];

    const float* qbase = xq + (size_t)b * HW * DIM;
    const float* vbase = xv + (size_t)b * HW * CH;

    // Per-thread LDS byte offsets for the async tile copies (16B per lane).
    const uint32_t qls0 = (uint32_t)(uintptr_t)(void*)&qs[0][0] + (uint32_t)tid * 16u;
    const uint32_t vls0 = (uint32_t)(uintptr_t)(void*)&vs[0][0] + (uint32_t)tid * 16u;

    // Issue a full tile prefetch (10 async b128 ops per thread: 4KB q + 16KB V).
    auto issue_tile = [&](int j0, int buf) {
        const char* qsrc = (const char*)(qbase + (size_t)j0 * DIM) + (size_t)tid * 16;
        const char* vsrc = (const char*)(vbase + (size_t)j0 * CH) + (size_t)tid * 16;
        const uint32_t qd = qls0 + (uint32_t)buf * (16 * DIM * 4);
        const uint32_t vd = vls0 + (uint32_t)buf * (16 * CH * 4);
#pragma unroll
        for (int r = 0; r < 2; ++r)
            async_copy16(qd + r * 2048u, qsrc + r * 2048);
#pragma unroll
        for (int r = 0; r < 8; ++r)
            async_copy16(vd + r * 2048u, vsrc + r * 2048);
    };

    // Preload B-fragments of phi_g^T (fixed per wave): B[k=d][n=i]
    // lane: N = lp -> row i0+lp ; step s: K = 4s + 2*half + {0,1}
    v2f bq[16];
    {
        const float* pgrow = pg + ((size_t)b * HW + i0 + lp) * DIM + 2 * half;
#pragma unroll
        for (int s = 0; s < 16; ++s)
            bq[s] = *(const v2f*)(pgrow + 4 * s);
    }

    v8f o[16];  // O accumulator: 16 rows x 256 cols fp32 (128 VGPRs)
    const v8f zero8 = {0.f, 0.f, 0.f, 0.f, 0.f, 0.f, 0.f, 0.f};
#pragma unroll
    for (int cc = 0; cc < 16; ++cc) o[cc] = zero8;

    float m = -INFINITY;  // running row max (row = lp, replicated in both halves)
    float l = 0.0f;       // running row sum

    issue_tile(0, 0);  // prologue prefetch of tile 0

#pragma unroll 1
    for (int t = 0; t < NTILES; ++t) {
        const int buf = t & 1;
        if (t + 1 < NTILES) {
            issue_tile((t + 1) * 16, buf ^ 1);  // prefetch next tile
            // 10 newest async ops are tile t+1; <=10 outstanding => tile t landed
            asm volatile("s_wait_asynccnt 10" ::: "memory");
        } else {
            asm volatile("s_wait_asynccnt 0" ::: "memory");
        }
        __syncthreads();  // all waves' tile-t data visible

        const float* qt = qs[buf];
        const float* vt = vs[buf];

        // ---- scores, transposed: T[j][i] = sum_d q[j0+j][d] * pg[i0+i][d]
        // A = q tile (M=j), fragments read straight from LDS; B = bq (preloaded)
        v8f c = zero8;
#pragma unroll
        for (int s = 0; s < 16; ++s) {
            v2f a = *(const v2f*)&qt[lp * DIM + 4 * s + 2 * half];
            c = __builtin_amdgcn_wmma_f32_16x16x4_f32(false, a, false, bq[s],
                                                      (short)0, c, false, false);
        }

        // ---- online softmax; lane lp tracks row i0+lp
        // c reg r holds T[j = r + 8*half][i = lp]
        float mt = c[0];
#pragma unroll
        for (int r = 1; r < 8; ++r) mt = fmaxf(mt, c[r]);
        mt = fmaxf(mt, __shfl_xor(mt, 16));
        const float mold = m;
        const float mnew = fmaxf(mold, mt);
        float p0 = __expf(c[0] - mnew), p1 = __expf(c[1] - mnew);
        float p2 = __expf(c[2] - mnew), p3 = __expf(c[3] - mnew);
        float p4 = __expf(c[4] - mnew), p5 = __expf(c[5] - mnew);
        float p6 = __expf(c[6] - mnew), p7 = __expf(c[7] - mnew);
        float ts = ((p0 + p1) + (p2 + p3)) + ((p4 + p5) + (p6 + p7));
        ts += __shfl_xor(ts, 16);
        const float alpha = __expf(mold - mnew);
        l = l * alpha + ts;
        m = mnew;

        // Rescale O only when the max moved anywhere in the wave (uniform branch)
        if (__ballot(mnew != mold) != 0ull) {
#pragma unroll
            for (int v = 0; v < 8; ++v) {
                const float alo = __shfl(alpha, v);
                const float ahi = __shfl(alpha, v + 8);
                const float bc = half ? ahi : alo;
#pragma unroll
                for (int cc = 0; cc < 16; ++cc) o[cc][v] *= bc;
            }
        }

        // ---- build PV A-fragments A[i][j] from transposed P tile
        // half-swap + select (derived from 16x16 C-layout <-> 16x4 A-layout)
        const float s0 = __shfl_xor(p0, 16), s1 = __shfl_xor(p1, 16);
        const float s2 = __shfl_xor(p2, 16), s3 = __shfl_xor(p3, 16);
        const float s4 = __shfl_xor(p4, 16), s5 = __shfl_xor(p5, 16);
        const float s6 = __shfl_xor(p6, 16), s7 = __shfl_xor(p7, 16);
        const v2f pa0 = {half ? s2 : p0, half ? s3 : p1};  // K = j 0..3
        const v2f pa1 = {half ? s6 : p4, half ? s7 : p5};  // K = j 4..7
        const v2f pa2 = {half ? p2 : s0, half ? p3 : s1};  // K = j 8..11
        const v2f pa3 = {half ? p6 : s4, half ? p7 : s5};  // K = j 12..15

        // ---- O[i][c] += P[i][j] * V[j][c], B-fragments from LDS V tile
#pragma unroll
        for (int cc = 0; cc < 16; ++cc) {
            const float* vcol = vt + (size_t)(2 * half) * CH + cc * 16 + lp;
            const v2f bv0 = {vcol[0 * CH], vcol[1 * CH]};
            o[cc] = __builtin_amdgcn_wmma_f32_16x16x4_f32(false, pa0, false, bv0,
                                                          (short)0, o[cc], false, false);
            const v2f bv1 = {vcol[4 * CH], vcol[5 * CH]};
            o[cc] = __builtin_amdgcn_wmma_f32_16x16x4_f32(false, pa1, false, bv1,
                                                          (short)0, o[cc], false, false);
            const v2f bv2 = {vcol[8 * CH], vcol[9 * CH]};
            o[cc] = __builtin_amdgcn_wmma_f32_16x16x4_f32(false, pa2, false, bv2,
                                                          (short)0, o[cc], false, false);
            const v2f bv3 = {vcol[12 * CH], vcol[13 * CH]};
            o[cc] = __builtin_amdgcn_wmma_f32_16x16x4_f32(false, pa3, false, bv3,
                                                          (short)0, o[cc], false, false);
        }

        __syncthreads();  // all waves done reading buf before it is overwritten
    }

    // ---- epilogue: out[b, i, c] = k * O/l + x   (first 256 channels)
    const float linv = 1.0f / l;
    const float* xrow = xres + (size_t)b * HW * CH;
    float* orow = out + (size_t)b * HW * (2 * CH);
#pragma unroll
    for (int v = 0; v < 8; ++v) {
        const float llo = __shfl(linv, v);
        const float lhi = __shfl(linv, v + 8);
        const float bc = half ? lhi : llo;
        const int row = i0 + v + 8 * half;
#pragma unroll
        for (int cc = 0; cc < 16; ++cc) {
            const int col = cc * 16 + lp;
            const float val = kk * o[cc][v] * bc + xrow[(size_t)row * CH + col];
            orow[(size_t)row * (2 * CH) + col] = val;
        }
    }
}

// out[b, i, 256:512] = g  (float4 copy)
__global__ __launch_bounds__(256) void concat_g_kernel(const float* __restrict__ g,
                                                       float* __restrict__ out) {
    const int idx = blockIdx.x * blockDim.x + threadIdx.x;  // float4 index
    const float4* g4 = (const float4*)g;
    float4* o4 = (float4*)out;
    const int row = idx >> 6;  // 64 float4 per 256-ch row
    const int c4 = idx & 63;
    o4[(size_t)row * 128 + 64 + c4] = g4[idx];
}

extern "C" void kernel_launch(void* const* d_in, const int* in_sizes, int n_in,
                              void* d_out, int out_size, void* d_ws, size_t ws_size,
                              hipStream_t stream) {
    const float* g  = (const float*)d_in[0];
    const float* x  = (const float*)d_in[1];
    const float* xq = (const float*)d_in[2];
    const float* pg = (const float*)d_in[3];
    const float* xv = (const float*)d_in[4];
    const float* k  = (const float*)d_in[5];
    float* out = (float*)d_out;

    dim3 grid(HW / ROWS_PER_WG, BATCH);
    attn_flash_kernel<<<grid, 32 * WAVES, 0, stream>>>(xq, pg, xv, x, k, out);

    const int n4 = BATCH * HW * CH / 4;  // 1,048,576 float4
    concat_g_kernel<<<n4 / 256, 256, 0, stream>>>(g, out);
}